// DiffPoolPrompt_74852690035344
// MI455X (gfx1250) — compile-verified
//
#include <hip/hip_runtime.h>

#define N_NODES 100000
#define N_EDGES 3200000
#define IN_CH   256
#define NK      10
#define NKP     16               // padded cluster dim for WMMA
#define NTILES  (N_NODES / 16)   // 6250, exact

typedef float v2f __attribute__((ext_vector_type(2)));
typedef float v8f __attribute__((ext_vector_type(8)));

// ---------------------------------------------------------------------------
// K0: cvec[n] = (sum_k ce[k][:]) @ W[:, n]   (the folded prompt-shift constant)
// one block of 256 threads
// ---------------------------------------------------------------------------
__global__ void k_precompute_c(const float* __restrict__ W,
                               const float* __restrict__ ce,
                               float* __restrict__ cvec) {
    __shared__ float csum[IN_CH];
    int t = threadIdx.x;
    float s = 0.f;
#pragma unroll
    for (int k = 0; k < NK; ++k) s += ce[k * IN_CH + t];
    csum[t] = s;
    __syncthreads();
    if (t < NK) {
        float acc = 0.f;
        for (int j = 0; j < IN_CH; ++j) acc += csum[j] * W[j * NK + t];
        cvec[t] = acc;
    }
}

// ---------------------------------------------------------------------------
// Degree (incl. self-loop) and dinv = rsqrt(deg)
// ---------------------------------------------------------------------------
__global__ void k_deg_init(float* __restrict__ deg) {
    int i = blockIdx.x * blockDim.x + threadIdx.x;
    if (i < N_NODES) deg[i] = 1.0f;   // self-loop
}

__global__ void k_deg_count(const long long* __restrict__ ei,
                            float* __restrict__ deg) {
    int e = blockIdx.x * blockDim.x + threadIdx.x;
    if (e < N_EDGES) {
        int d = (int)ei[N_EDGES + e];   // dst row
        atomicAdd(&deg[d], 1.0f);
    }
}

__global__ void k_dinv(float* __restrict__ deg) {
    int i = blockIdx.x * blockDim.x + threadIdx.x;
    if (i < N_NODES) deg[i] = rsqrtf(deg[i]);   // deg >= 1 always
}

// ---------------------------------------------------------------------------
// K3: hW = x @ W + c  via v_wmma_f32_16x16x4_f32, one wave per 16-row tile.
//     W is staged zero-padded (256 x 16) in LDS so B-fragment loads are
//     unconditional ds_load_b32 (bank-conflict-free), no divergent exec.
//     Also seeds logits with the self-loop term hW * dinv^2.
// A frag (16x4 f32): lane l<16 -> row=l,  K = kt+{0,1}; lane>=16 -> K = kt+{2,3}
// B frag (4x16 f32): lane holds col=lane&15, same K split across lane halves
// C/D (16x16 f32)  : vgpr v -> M = v + 8*(lane>=16), N = lane&15
// ---------------------------------------------------------------------------
__global__ void __launch_bounds__(256)
k_gemm_xw(const float* __restrict__ x,
          const float* __restrict__ W,
          const float* __restrict__ cvec,
          const float* __restrict__ dinv,
          float* __restrict__ hW,
          float* __restrict__ logits) {
    __shared__ float sW[IN_CH * NKP];          // 16 KB, cols 10..15 zeroed

    // cooperative zero-padded stage of W: 4096 entries / 256 threads
    for (int i = threadIdx.x; i < IN_CH * NKP; i += 256) {
        int k = i >> 4, c = i & 15;
        sW[i] = (c < NK) ? W[k * NK + c] : 0.f;
    }
    __syncthreads();                            // all waves participate

    const int lane = threadIdx.x & 31;
    const int wave = threadIdx.x >> 5;
    const int tile = blockIdx.x * 8 + wave;
    if (tile >= NTILES) return;                 // wave-uniform after barrier

    const int row0  = tile * 16;
    const int col   = lane & 15;
    const int khalf = (lane >> 4) * 2;          // 0 or 2

    const float cinit = (col < NK) ? cvec[col] : 0.f;
    v8f acc;
#pragma unroll
    for (int i = 0; i < 8; ++i) acc[i] = cinit;

    const float* xrow = x + (size_t)(row0 + col) * IN_CH;
    const float* sWl  = sW + khalf * NKP + col; // lane's B base

#pragma unroll 8
    for (int kt = 0; kt < IN_CH; kt += 4) {
        v2f a = *(const v2f*)(xrow + kt + khalf);   // 8B-aligned global_load_b64
        v2f b;
        b[0] = sWl[kt * NKP];                       // ds_load_b32, no divergence
        b[1] = sWl[kt * NKP + NKP];
        acc = __builtin_amdgcn_wmma_f32_16x16x4_f32(
            /*neg_a=*/false, a, /*neg_b=*/false, b,
            /*c_mod=*/(short)0, acc, /*reuse_a=*/false, /*reuse_b=*/false);
    }

#pragma unroll
    for (int v = 0; v < 8; ++v) {
        int r = row0 + v + 8 * (lane >> 4);
        if (col < NK) {
            float val = acc[v];
            hW[r * NK + col] = val;
            float di = dinv[r];
            logits[r * NK + col] = val * di * di;   // self-loop seed
        }
    }
}

// ---------------------------------------------------------------------------
// K4: edge scatter: logits[d] += hW[s] * dinv[s]*dinv[d]   (L2-resident atomics)
// ---------------------------------------------------------------------------
__global__ void k_edge_scatter(const long long* __restrict__ ei,
                               const float* __restrict__ dinv,
                               const float* __restrict__ hW,
                               float* __restrict__ logits) {
    int e = blockIdx.x * blockDim.x + threadIdx.x;
    if (e >= N_EDGES) return;
    int s = (int)ei[e];
    int d = (int)ei[N_EDGES + e];
    float nrm = dinv[s] * dinv[d];
    const float* hs = hW + (size_t)s * NK;
    float* ld = logits + (size_t)d * NK;
    float m[NK];
#pragma unroll
    for (int c = 0; c < NK; ++c) m[c] = hs[c] * nrm;
#pragma unroll
    for (int c = 0; c < NK; ++c) atomicAdd(&ld[c], m[c]);
}

// ---------------------------------------------------------------------------
// K5: per-row softmax(logits+bias), p = s @ ce, out = x + p. One wave per row.
// ---------------------------------------------------------------------------
__global__ void k_softmax_out(const float* __restrict__ x,
                              const float* __restrict__ bias,
                              const float* __restrict__ ce,
                              const float* __restrict__ logits,
                              float* __restrict__ out) {
    const int lane = threadIdx.x & 31;
    const int wave = threadIdx.x >> 5;
    const int row  = blockIdx.x * 8 + wave;
    if (row >= N_NODES) return;

    float l[NK];
    float mx = -3.0e38f;
#pragma unroll
    for (int k = 0; k < NK; ++k) {
        l[k] = logits[row * NK + k] + bias[k];
        mx = fmaxf(mx, l[k]);
    }
    float sum = 0.f;
#pragma unroll
    for (int k = 0; k < NK; ++k) { l[k] = __expf(l[k] - mx); sum += l[k]; }
    const float inv = 1.f / sum;
#pragma unroll
    for (int k = 0; k < NK; ++k) l[k] *= inv;

#pragma unroll
    for (int jj = 0; jj < IN_CH / 32; ++jj) {
        int j = lane + jj * 32;
        float p = 0.f;
#pragma unroll
        for (int k = 0; k < NK; ++k) p = fmaf(l[k], ce[k * IN_CH + j], p);
        out[(size_t)row * IN_CH + j] = x[(size_t)row * IN_CH + j] + p;
    }
}

// ---------------------------------------------------------------------------
extern "C" void kernel_launch(void* const* d_in, const int* in_sizes, int n_in,
                              void* d_out, int out_size, void* d_ws, size_t ws_size,
                              hipStream_t stream) {
    (void)in_sizes; (void)n_in; (void)out_size; (void)ws_size;
    const float*     x    = (const float*)d_in[0];
    const long long* ei   = (const long long*)d_in[1];   // int64 [2, E]
    /* d_in[2] = batch (unused) */
    const float*     W    = (const float*)d_in[3];       // [256, 10]
    const float*     bias = (const float*)d_in[4];       // [10]
    const float*     ce   = (const float*)d_in[5];       // [10, 256]
    float*           out  = (float*)d_out;

    // workspace layout (floats)
    float* wsF    = (float*)d_ws;
    float* deg    = wsF;                        // N  (becomes dinv in place)
    float* cvec   = wsF + N_NODES;              // 16
    float* hW     = cvec + 16;                  // N*10
    float* logits = hW + (size_t)N_NODES * NK;  // N*10

    const int TB = 256;
    k_precompute_c<<<1, TB, 0, stream>>>(W, ce, cvec);
    k_deg_init    <<<(N_NODES + TB - 1) / TB, TB, 0, stream>>>(deg);
    k_deg_count   <<<(N_EDGES + TB - 1) / TB, TB, 0, stream>>>(ei, deg);
    k_dinv        <<<(N_NODES + TB - 1) / TB, TB, 0, stream>>>(deg);
    k_gemm_xw     <<<(NTILES + 7) / 8, TB, 0, stream>>>(x, W, cvec, deg, hW, logits);
    k_edge_scatter<<<(N_EDGES + TB - 1) / TB, TB, 0, stream>>>(ei, deg, hW, logits);
    k_softmax_out <<<(N_NODES + 7) / 8, TB, 0, stream>>>(x, bias, ce, logits, out);
}